// HungarianCELoss_55018531062741
// MI455X (gfx1250) — compile-verified
//
#include <hip/hip_runtime.h>
#include <hip/hip_bf16.h>

// Problem constants (B=64, K=5, H=W=320)
#define BATCH 64
#define KSLOT 5
#define HW    102400            // 320*320
#define NV4   (HW / 4)          // 25600 float4 per (b,k) row
#define BLOCKS_PER_B 50         // 50*256 threads -> exactly 2 float4 iters/thread
#define THREADS 256
#define NQ 11                   // per-batch sums: S[0..4], I[0..4], Tfg
#define EPS_F 1e-6f

#ifndef __has_builtin
#define __has_builtin(x) 0
#endif

#if __has_builtin(__builtin_amdgcn_wmma_f32_16x16x4_f32)
#define HAVE_WMMA4 1
#else
#define HAVE_WMMA4 0
#endif

typedef float v2f __attribute__((ext_vector_type(2)));
typedef float v8f __attribute__((ext_vector_type(8)));

// Full 32-lane sum, result valid in every lane.
// CDNA5 path: one V_WMMA_F32_16X16X4_F32 with B = ones.
//   A layout (16x4 f32): lane L<16 holds A[L,0..1], lane L>=16 holds A[L-16,2..3].
//   We place v in the first K slot of each lane -> A[m,0]=v_m, A[m,2]=v_{m+16}.
//   D[m,n] = v_m + v_{m+16} for all n. C/D layout: lane L<16 holds D[0..7,L],
//   lane L>=16 holds D[8..15,L-16]. Summing the 8 regs + shfl_xor(16) gives
//   sum over all 32 lanes in every lane.
__device__ __forceinline__ float wave_sum32(float v) {
#if HAVE_WMMA4
  v2f a;  a[0] = v;    a[1] = 0.0f;
  v2f b1; b1[0] = 1.0f; b1[1] = 1.0f;
  v8f c = {};
  c = __builtin_amdgcn_wmma_f32_16x16x4_f32(
      /*neg_a=*/false, a, /*neg_b=*/false, b1,
      /*c_mod=*/(short)0, c, /*reuse_a=*/false, /*reuse_b=*/false);
  float s = ((c[0] + c[1]) + (c[2] + c[3])) + ((c[4] + c[5]) + (c[6] + c[7]));
  s += __shfl_xor(s, 16);
  return s;
#else
  for (int off = 16; off > 0; off >>= 1) v += __shfl_xor(v, off);
  return v;
#endif
}

__device__ __forceinline__ float fast_sigmoid(float x) {
  return __builtin_amdgcn_rcpf(1.0f + __expf(-x));
}

__device__ __forceinline__ float bce_term(float x, float a) {
  // max(x,0) - x*a + log1p(exp(-|x|)); log(1+y) abs-err is negligible at this scale
  return fmaxf(x, 0.0f) - x * a + __logf(1.0f + __expf(-fabsf(x)));
}

// -------- Pass 1: per-batch sums S_k, I_k (= <sigmoid(x_k), fg>), Tfg --------
__global__ void __launch_bounds__(THREADS)
pass1_kernel(const float* __restrict__ slot, const float* __restrict__ tgt,
             float* __restrict__ sums) {
  const int b = blockIdx.y;
  const float4* __restrict__ t4 = (const float4*)(tgt + (size_t)b * HW);
  const float4* __restrict__ s4 = (const float4*)(slot + (size_t)b * KSLOT * HW);

  float accS[KSLOT] = {0.f, 0.f, 0.f, 0.f, 0.f};
  float accI[KSLOT] = {0.f, 0.f, 0.f, 0.f, 0.f};
  float accT = 0.0f;

  for (int i = blockIdx.x * THREADS + threadIdx.x; i < NV4;
       i += BLOCKS_PER_B * THREADS) {
    const float4 f = t4[i];
    accT += (f.x + f.y) + (f.z + f.w);
#pragma unroll
    for (int k = 0; k < KSLOT; ++k) {
      const float4 x = s4[(size_t)k * NV4 + i];
      const float px = fast_sigmoid(x.x);
      const float py = fast_sigmoid(x.y);
      const float pz = fast_sigmoid(x.z);
      const float pw = fast_sigmoid(x.w);
      accS[k] += (px + py) + (pz + pw);
      accI[k] += px * f.x + py * f.y + pz * f.z + pw * f.w;
    }
  }

  float q[NQ];
#pragma unroll
  for (int k = 0; k < KSLOT; ++k) { q[k] = accS[k]; q[KSLOT + k] = accI[k]; }
  q[10] = accT;

  __shared__ float red[(THREADS / 32) * NQ];
  const int wave = threadIdx.x >> 5;
  const int lane = threadIdx.x & 31;
#pragma unroll
  for (int i = 0; i < NQ; ++i) {
    const float s = wave_sum32(q[i]);       // uniform: all lanes active (EXEC=~0)
    if (lane == 0) red[wave * NQ + i] = s;
  }
  __syncthreads();
  if (threadIdx.x < NQ) {
    float s = 0.0f;
#pragma unroll
    for (int w = 0; w < THREADS / 32; ++w) s += red[w * NQ + threadIdx.x];
    atomicAdd(&sums[b * NQ + threadIdx.x], s);
  }
}

// -------- Pick k0[b]: argmin_k (cost_fg(k) - cost_bg(k)), ties -> smallest k --------
__global__ void pick_kernel(const float* __restrict__ sums, int* __restrict__ k0) {
  const int b = threadIdx.x;
  if (b >= BATCH) return;
  const float* s = sums + b * NQ;
  const float Tf = s[10];
  const float Tb = (float)HW - Tf;
  float best = 3.0e38f;
  int bk = 0;
#pragma unroll
  for (int k = 0; k < KSLOT; ++k) {
    const float S = s[k];
    const float I = s[KSLOT + k];
    const float iB = S - I;
    const float uF = S + Tf - I + EPS_F;
    const float uB = S + Tb - iB + EPS_F;
    const float d = (1.0f - I / uF) - (1.0f - iB / uB);
    if (d < best) { best = d; bk = k; }   // strict '<' => first-min tie-break
  }
  k0[b] = bk;
}

// -------- Pass 2: masked BCE sum (slot_logits re-read mostly from 192MB L2) --------
__global__ void __launch_bounds__(THREADS)
pass2_kernel(const float* __restrict__ slot, const float* __restrict__ tgt,
             const int* __restrict__ k0, double* __restrict__ bce_acc) {
  const int b = blockIdx.y;
  const int kfg = k0[b];
  const float4* __restrict__ t4 = (const float4*)(tgt + (size_t)b * HW);
  const float4* __restrict__ s4 = (const float4*)(slot + (size_t)b * KSLOT * HW);

  float acc = 0.0f;
  for (int i = blockIdx.x * THREADS + threadIdx.x; i < NV4;
       i += BLOCKS_PER_B * THREADS) {
    const float4 f = t4[i];
#pragma unroll
    for (int k = 0; k < KSLOT; ++k) {
      const float4 x = s4[(size_t)k * NV4 + i];
      const bool isfg = (k == kfg);
      const float ax = isfg ? f.x : 1.0f - f.x;
      const float ay = isfg ? f.y : 1.0f - f.y;
      const float az = isfg ? f.z : 1.0f - f.z;
      const float aw = isfg ? f.w : 1.0f - f.w;
      acc += (bce_term(x.x, ax) + bce_term(x.y, ay)) +
             (bce_term(x.z, az) + bce_term(x.w, aw));
    }
  }

  const float wsum = wave_sum32(acc);
  __shared__ float red[THREADS / 32];
  const int wave = threadIdx.x >> 5;
  const int lane = threadIdx.x & 31;
  if (lane == 0) red[wave] = wsum;
  __syncthreads();
  if (threadIdx.x == 0) {
    float s = 0.0f;
#pragma unroll
    for (int w = 0; w < THREADS / 32; ++w) s += red[w];
    atomicAdd(bce_acc, (double)s);
  }
}

__global__ void final_kernel(const double* __restrict__ bce_acc,
                             float* __restrict__ out) {
  out[0] = (float)(bce_acc[0] * (1.0 / ((double)BATCH * KSLOT * HW)));
}

extern "C" void kernel_launch(void* const* d_in, const int* in_sizes, int n_in,
                              void* d_out, int out_size, void* d_ws, size_t ws_size,
                              hipStream_t stream) {
  (void)in_sizes; (void)n_in; (void)out_size; (void)ws_size;
  // setup_inputs order: [0]=fg_logits (UNUSED by reference), [1]=slot_logits, [2]=target
  const float* slot = (const float*)d_in[1];
  const float* tgt  = (const float*)d_in[2];

  double* bce_acc = (double*)d_ws;                                    // 8 B
  float*  sums    = (float*)((char*)d_ws + sizeof(double));           // 64*11 f32
  int*    k0      = (int*)((char*)d_ws + sizeof(double) +
                           (size_t)BATCH * NQ * sizeof(float));       // 64 i32

  hipMemsetAsync(d_ws, 0, sizeof(double) + (size_t)BATCH * NQ * sizeof(float),
                 stream);

  dim3 grid(BLOCKS_PER_B, BATCH);
  pass1_kernel<<<grid, THREADS, 0, stream>>>(slot, tgt, sums);
  pick_kernel<<<1, 64, 0, stream>>>(sums, k0);
  pass2_kernel<<<grid, THREADS, 0, stream>>>(slot, tgt, k0, bce_acc);
  final_kernel<<<1, 1, 0, stream>>>(bce_acc, (float*)d_out);
}